// TimeEncodings_41437844471912
// MI455X (gfx1250) — compile-verified
//
#include <hip/hip_runtime.h>
#include <hip/hip_bf16.h>
#include <stdint.h>

#define AS1 __attribute__((address_space(1)))
#define AS3 __attribute__((address_space(3)))

#ifndef __has_builtin
#define __has_builtin(x) 0
#endif

#if defined(__HIP_DEVICE_COMPILE__) && defined(__gfx1250__) &&                 \
    __has_builtin(__builtin_amdgcn_global_load_async_to_lds_b128) &&           \
    __has_builtin(__builtin_amdgcn_global_store_async_from_lds_b128) &&        \
    __has_builtin(__builtin_amdgcn_s_wait_asynccnt)
#define USE_CDNA5_ASYNC 1
#else
#define USE_CDNA5_ASYNC 0
#endif

// Native clang vector types (NOT HIP_vector_type classes) so the
// nontemporal/async builtins accept pointers to them.
typedef float fv4 __attribute__((ext_vector_type(4)));
typedef int   v4i __attribute__((ext_vector_type(4)));

namespace {
constexpr int kThreads      = 256;                 // 8 wave32s per block
constexpr int kTiles        = 8;                   // 8 x 16B per thread
constexpr int kVec4PerBlock = kThreads * kTiles;   // 2048 fv4 = 32 KiB/block
}

#if USE_CDNA5_ASYNC
__device__ __forceinline__ AS3 v4i* to_lds(void* p) {
  // Generic LDS pointers carry the LDS byte offset in the low 32 bits
  // (flat-aperture rule); truncation == addrspacecast.
  return (AS3 v4i*)(uint32_t)(uintptr_t)p;
}
__device__ __forceinline__ AS1 v4i* to_gbl(const void* p) {
  return (AS1 v4i*)(uintptr_t)p;
}
#endif

// Bulk identity copy: each block moves 32 KiB, fully coalesced b128 accesses.
// CDNA5 path: stage through LDS with the gfx1250 async DMA instructions
// (GLOBAL_LOAD_ASYNC_TO_LDS_B128 / GLOBAL_STORE_ASYNC_FROM_LDS_B128,
// synchronized with s_wait_asynccnt). Fallback: non-temporal fv4 copy.
__global__ __launch_bounds__(kThreads)
void te_identity_copy_main(const fv4* __restrict__ src,
                           fv4* __restrict__ dst) {
  const unsigned t    = threadIdx.x;
  const size_t   base = (size_t)blockIdx.x * (size_t)kVec4PerBlock;
#if USE_CDNA5_ASYNC
  __shared__ fv4 stage[kVec4PerBlock];
  const fv4* s = src + base;
  fv4*       d = dst + base;
#pragma unroll
  for (int j = 0; j < kTiles; ++j) {
    const unsigned slot = j * kThreads + t;
    __builtin_amdgcn_global_load_async_to_lds_b128(
        to_gbl(s + slot), to_lds(&stage[slot]), /*offset=*/0, /*cpol=*/0);
  }
  // Wait for all 8 in-flight async loads of this wave before reading LDS back.
  __builtin_amdgcn_s_wait_asynccnt(0);
#pragma unroll
  for (int j = 0; j < kTiles; ++j) {
    const unsigned slot = j * kThreads + t;
    __builtin_amdgcn_global_store_async_from_lds_b128(
        to_gbl(d + slot), to_lds(&stage[slot]), /*offset=*/0, /*cpol=*/0);
  }
  // Drain stores (s_endpgm would also wait-idle; explicit for clarity).
  __builtin_amdgcn_s_wait_asynccnt(0);
#else
#pragma unroll
  for (int j = 0; j < kTiles; ++j) {
    const size_t i = base + (size_t)(j * kThreads + t);
    fv4 v = __builtin_nontemporal_load(&src[i]);
    __builtin_nontemporal_store(v, &dst[i]);
  }
#endif
}

// Generic scalar tail for sizes that are not a multiple of 32 KiB (zero here).
__global__ __launch_bounds__(kThreads)
void te_identity_copy_tail(const float* __restrict__ src,
                           float* __restrict__ dst,
                           size_t start, size_t n) {
  size_t i = start + (size_t)blockIdx.x * blockDim.x + threadIdx.x;
  if (i < n) dst[i] = src[i];
}

extern "C" void kernel_launch(void* const* d_in, const int* in_sizes, int n_in,
                              void* d_out, int out_size, void* d_ws, size_t ws_size,
                              hipStream_t stream) {
  (void)n_in; (void)d_ws; (void)ws_size; (void)in_sizes;
  const float* X   = (const float*)d_in[0];   // d_in[1] ("times") is dead in the reference
  float*       out = (float*)d_out;

  const size_t n          = (size_t)out_size;          // fp32 elements (8192*4096)
  const size_t n4         = n / 4;                     // fv4 elements
  const size_t fullBlocks = n4 / kVec4PerBlock;        // 4096 for this shape

  if (fullBlocks) {
    te_identity_copy_main<<<(unsigned)fullBlocks, kThreads, 0, stream>>>(
        (const fv4*)X, (fv4*)out);
  }
  const size_t done = fullBlocks * (size_t)kVec4PerBlock * 4;
  const size_t rem  = n - done;
  if (rem) {
    unsigned blocks = (unsigned)((rem + kThreads - 1) / kThreads);
    te_identity_copy_tail<<<blocks, kThreads, 0, stream>>>(X, out, done, n);
  }
}